// AttnBlock_86577950753016
// MI455X (gfx1250) — compile-verified
//
#include <hip/hip_runtime.h>
#include <hip/hip_bf16.h>
#include <math.h>

// ---------------------------------------------------------------------------
// Decode-step transformer block for MI455X (gfx1250, wave32).
// Memory-bound: ~158 MB of fp32 weights+KV per call -> ~6.8us floor @23.3TB/s.
// fp32 kept end-to-end (bandwidth-bound; matrix engine idle either way).
// WMMA (V_WMMA_F32_16X16X4_F32) used where real matrix shape exists:
// scores = Qpad(16x256) @ K^T(256x8192), fp32 accumulate (exact precision).
// ---------------------------------------------------------------------------

#define TPB 256
#define HEAD_DIM 256
#define HIDDEN 1536
#define INTER 6144
#define NQH 8
#define PAST_LEN 8191
#define S_TOTAL 8192
#define ATTN_SCALE 0.0625f /* 256^-0.5 */

typedef __attribute__((ext_vector_type(2))) float v2f;
typedef __attribute__((ext_vector_type(8))) float v8f;

// ------------------- split-K GEMV: part[chunk][col] = sum_k x[k]*W[k][col] ---
// W is (K, N) row-major => consecutive threads read consecutive addresses.
__global__ __launch_bounds__(TPB) void gemv_partial(
    const float* __restrict__ W, const float* __restrict__ x,
    int K, int N, int kchunk, float* __restrict__ part)
{
  extern __shared__ float xs[];
  const int col = blockIdx.x * TPB + threadIdx.x;
  const int k0  = blockIdx.y * kchunk;
  int len = K - k0; if (len > kchunk) len = kchunk;
  for (int i = threadIdx.x; i < len; i += TPB) xs[i] = x[k0 + i];
  __syncthreads();
  float acc = 0.0f;
  if (col < N) {
    const float* Wp = W + (size_t)k0 * N + col;
    #pragma unroll 4
    for (int i = 0; i < len; ++i) acc = fmaf(xs[i], Wp[(size_t)i * N], acc);
    part[(size_t)blockIdx.y * N + col] = acc;
  }
}

__global__ __launch_bounds__(TPB) void reduce_bias(
    const float* __restrict__ part, const float* __restrict__ bias,
    int N, int nc, float* __restrict__ out)
{
  int j = blockIdx.x * TPB + threadIdx.x;
  if (j >= N) return;
  float s = bias[j];
  for (int c = 0; c < nc; ++c) s += part[(size_t)c * N + j];
  out[j] = s;
}

__global__ __launch_bounds__(TPB) void reduce_plain(
    const float* __restrict__ part, int N, int nc, float* __restrict__ out)
{
  int j = blockIdx.x * TPB + threadIdx.x;
  if (j >= N) return;
  float s = 0.0f;
  for (int c = 0; c < nc; ++c) s += part[(size_t)c * N + j];
  out[j] = s;
}

__global__ __launch_bounds__(TPB) void reduce_bias_residual(
    const float* __restrict__ part, const float* __restrict__ bias,
    const float* __restrict__ resid, int N, int nc, float* __restrict__ out)
{
  int j = blockIdx.x * TPB + threadIdx.x;
  if (j >= N) return;
  float s = bias[j] + resid[j];
  for (int c = 0; c < nc; ++c) s += part[(size_t)c * N + j];
  out[j] = s;
}

// gu = GELU_exact(g) * u  with g = sum(partG)+bg, u = sum(partU)+bu
__global__ __launch_bounds__(TPB) void reduce_geluup(
    const float* __restrict__ partG, const float* __restrict__ partU,
    const float* __restrict__ bg, const float* __restrict__ bu,
    int N, int nc, float* __restrict__ out)
{
  int j = blockIdx.x * TPB + threadIdx.x;
  if (j >= N) return;
  float g = bg[j], u = bu[j];
  for (int c = 0; c < nc; ++c) { g += partG[(size_t)c * N + j]; u += partU[(size_t)c * N + j]; }
  float gelu = 0.5f * g * (1.0f + erff(g * 0.70710678118654752f)); // exact erf GELU
  out[j] = gelu * u;
}

// ------------------- attention scores via V_WMMA_F32_16X16X4_F32 ------------
// scores(8x8192) = Qpad(16x256) @ K^T ; one wave handles a 16-wide s tile.
// A layout (16x4 f32): lanes 0-15 M rows, lane half selects K pair {0,1}/{2,3}.
// B layout (4x16 f32): lanes 0-15 N cols (rows K=0/1 in v.x/v.y), half=rows 2/3.
// C layout: lanes 0-15 N=lane, VGPR r = M=r  -> heads 0..7 stored directly.
__global__ __launch_bounds__(TPB) void attn_scores_wmma(
    const float* __restrict__ q,       // 8*256, head-major
    const float* __restrict__ k_past,  // 8191 x 256
    const float* __restrict__ k_new,   // 256
    float* __restrict__ scores)        // 8 x 8192
{
  __shared__ float ldsQ[16 * HEAD_DIM];        // zero-padded to 16 rows
  for (int i = threadIdx.x; i < 16 * HEAD_DIM; i += TPB)
    ldsQ[i] = (i < NQH * HEAD_DIM) ? q[i] : 0.0f;
  __syncthreads();

  const int wave = threadIdx.x >> 5;           // 0..7
  const int lane = threadIdx.x & 31;
  const int s0   = (blockIdx.x * 8 + wave) * 16;
  const int rc   = lane & 15;                  // A: M row / B: N col
  const int kph  = (lane >> 4) << 1;           // K pair offset: 0 or 2

  const int s = s0 + rc;
  const float* krow = (s < PAST_LEN) ? (k_past + (size_t)s * HEAD_DIM) : k_new;
  const float* qrow = ldsQ + rc * HEAD_DIM;

  v8f c = {0.f, 0.f, 0.f, 0.f, 0.f, 0.f, 0.f, 0.f};
  #pragma unroll 8
  for (int kk = 0; kk < HEAD_DIM; kk += 4) {
    v2f a, b;
    a.x = qrow[kk + kph];  a.y = qrow[kk + kph + 1];
    b.x = krow[kk + kph];  b.y = krow[kk + kph + 1];
    c = __builtin_amdgcn_wmma_f32_16x16x4_f32(false, a, false, b,
                                              (short)0, c, false, false);
  }
  if (lane < 16) {
    const int ss = s0 + lane;
    #pragma unroll
    for (int r = 0; r < NQH; ++r)
      scores[(size_t)r * S_TOTAL + ss] = c[r] * ATTN_SCALE;
  }
}

// ------------------- per-head softmax over 8192 (in place) ------------------
__global__ __launch_bounds__(TPB) void softmax_head(float* __restrict__ scores)
{
  float* p = scores + (size_t)blockIdx.x * S_TOTAL;
  __shared__ float red[TPB];
  const int t = threadIdx.x;

  float m = -INFINITY;
  for (int i = t; i < S_TOTAL; i += TPB) m = fmaxf(m, p[i]);
  red[t] = m; __syncthreads();
  for (int w = TPB >> 1; w > 0; w >>= 1) {
    if (t < w) red[t] = fmaxf(red[t], red[t + w]);
    __syncthreads();
  }
  m = red[0]; __syncthreads();

  float sum = 0.0f;
  for (int i = t; i < S_TOTAL; i += TPB) { float e = expf(p[i] - m); p[i] = e; sum += e; }
  red[t] = sum; __syncthreads();
  for (int w = TPB >> 1; w > 0; w >>= 1) {
    if (t < w) red[t] += red[t + w];
    __syncthreads();
  }
  const float inv = 1.0f / red[0];
  for (int i = t; i < S_TOTAL; i += TPB) p[i] *= inv;
}

// ------------------- attn @ V : partials over 32 s-chunks -------------------
// V row (1KB) read once per chunk, fanned out to 8 head accumulators.
__global__ __launch_bounds__(TPB) void attn_av_partial(
    const float* __restrict__ attn,   // 8 x 8192 (normalized)
    const float* __restrict__ v_past, // 8191 x 256
    const float* __restrict__ v_new,  // 256
    float* __restrict__ part)         // 32 x 8 x 256
{
  __shared__ float ldsA[NQH * TPB];
  const int s0 = blockIdx.x * TPB;
  for (int i = threadIdx.x; i < NQH * TPB; i += TPB) {
    int h = i >> 8, so = i & 255;
    ldsA[i] = attn[(size_t)h * S_TOTAL + s0 + so];
  }
  __syncthreads();

  const int d = threadIdx.x;
  float acc[NQH];
  #pragma unroll
  for (int h = 0; h < NQH; ++h) acc[h] = 0.0f;

  for (int so = 0; so < TPB; ++so) {
    const int s = s0 + so;
    const float vv = (s < PAST_LEN) ? v_past[(size_t)s * HEAD_DIM + d] : v_new[d];
    #pragma unroll
    for (int h = 0; h < NQH; ++h) acc[h] = fmaf(ldsA[h * TPB + so], vv, acc[h]);
  }
  #pragma unroll
  for (int h = 0; h < NQH; ++h)
    part[((size_t)blockIdx.x * NQH + h) * HEAD_DIM + d] = acc[h];
}

// ---------------------------------------------------------------------------
extern "C" void kernel_launch(void* const* d_in, const int* in_sizes, int n_in,
                              void* d_out, int out_size, void* d_ws, size_t ws_size,
                              hipStream_t stream) {
  const float* x      = (const float*)d_in[0];
  const float* k_past = (const float*)d_in[1];
  const float* v_past = (const float*)d_in[2];
  const float* Wq = (const float*)d_in[3];  const float* bq = (const float*)d_in[4];
  const float* Wk = (const float*)d_in[5];  const float* bk = (const float*)d_in[6];
  const float* Wv = (const float*)d_in[7];  const float* bv = (const float*)d_in[8];
  const float* Wo = (const float*)d_in[9];  const float* bo = (const float*)d_in[10];
  const float* Wg = (const float*)d_in[11]; const float* bg = (const float*)d_in[12];
  const float* Wu = (const float*)d_in[13]; const float* bu = (const float*)d_in[14];
  const float* Wd = (const float*)d_in[15]; const float* bd = (const float*)d_in[16];
  float* out = (float*)d_out;

  float* ws = (float*)d_ws;
  float* q        = ws;                 // 2048
  float* k_new    = ws + 2048;          // 256
  float* v_new    = ws + 2304;          // 256
  float* scores   = ws + 2560;          // 8*8192 = 65536
  float* attn_out = ws + 68096;         // 2048
  float* x2       = ws + 70144;         // 1536
  float* gu       = ws + 71680;         // 6144
  float* part     = ws + 77824;         // up to 65536 (phases reuse it)
  float* partK    = part + 8192;        // QKV phase sub-buffers
  float* partV    = part + 8704;
  float* partU    = part + 12288;       // GU phase sub-buffer

  dim3 blk(TPB);

  // ---- QKV projections (split-K=4, coalesced row-major GEMV) ----
  gemv_partial<<<dim3(8, 4), blk, 384 * 4, stream>>>(Wq, x, HIDDEN, 2048, 384, part);
  gemv_partial<<<dim3(1, 4), blk, 384 * 4, stream>>>(Wk, x, HIDDEN, 256, 384, partK);
  gemv_partial<<<dim3(1, 4), blk, 384 * 4, stream>>>(Wv, x, HIDDEN, 256, 384, partV);
  reduce_bias<<<dim3(8), blk, 0, stream>>>(part,  bq, 2048, 4, q);
  reduce_bias<<<dim3(1), blk, 0, stream>>>(partK, bk, 256, 4, k_new);
  reduce_bias<<<dim3(1), blk, 0, stream>>>(partV, bv, 256, 4, v_new);

  // ---- attention scores via WMMA f32 16x16x4, then softmax ----
  attn_scores_wmma<<<dim3(64), blk, 0, stream>>>(q, k_past, k_new, scores);
  softmax_head<<<dim3(NQH), blk, 0, stream>>>(scores);

  // ---- attn @ V (32 s-chunk partials, deterministic reduce) ----
  attn_av_partial<<<dim3(32), blk, 0, stream>>>(scores, v_past, v_new, part);
  reduce_plain<<<dim3(8), blk, 0, stream>>>(part, 2048, 32, attn_out);

  // ---- O projection + residual ----
  gemv_partial<<<dim3(6, 4), blk, 512 * 4, stream>>>(Wo, attn_out, 2048, HIDDEN, 512, part);
  reduce_bias_residual<<<dim3(6), blk, 0, stream>>>(part, bo, x, HIDDEN, 4, x2);

  // ---- MLP: gate/up GEMVs -> exact GELU(g)*u ----
  gemv_partial<<<dim3(24, 2), blk, 768 * 4, stream>>>(Wg, x2, HIDDEN, INTER, 768, part);
  gemv_partial<<<dim3(24, 2), blk, 768 * 4, stream>>>(Wu, x2, HIDDEN, INTER, 768, partU);
  reduce_geluup<<<dim3(24), blk, 0, stream>>>(part, partU, bg, bu, INTER, 2, gu);

  // ---- down projection + residual -> output ----
  gemv_partial<<<dim3(6, 8), blk, 768 * 4, stream>>>(Wd, gu, INTER, HIDDEN, 768, part);
  reduce_bias_residual<<<dim3(6), blk, 0, stream>>>(part, bd, x2, HIDDEN, 8, out);
}